// GATLayer_11424613007836
// MI455X (gfx1250) — compile-verified
//
#include <hip/hip_runtime.h>

#define N_NODES 100000
#define N_EDGES 1600000
#define IN_DIM  128
#define OUT_DIM 64

typedef __attribute__((ext_vector_type(2))) float v2f;
typedef __attribute__((ext_vector_type(8))) float v8f;

// ---------- helpers ----------
__device__ __forceinline__ void atomAddF32(float* p, float v) {
    // relaxed agent-scope fp32 atomic -> global_atomic_add_f32 (no CAS loop)
    __hip_atomic_fetch_add(p, v, __ATOMIC_RELAXED, __HIP_MEMORY_SCOPE_AGENT);
}

// order-preserving float <-> uint mapping so segment-max can use u32 atomicMax
__device__ __forceinline__ unsigned enc_f(float f) {
    unsigned u = __float_as_uint(f);
    return (u & 0x80000000u) ? ~u : (u | 0x80000000u);
}
__device__ __forceinline__ float dec_f(unsigned u) {
    unsigned b = (u & 0x80000000u) ? (u & 0x7FFFFFFFu) : ~u;
    return __uint_as_float(b);
}
#define ENC_NEG_INF 0x007FFFFFu   // enc(-inf)

// ---------- kernel 1: init out / m_enc / denom ----------
__global__ void init_kernel(float* __restrict__ out, unsigned* __restrict__ m_enc,
                            float* __restrict__ denom) {
    long stride = (long)gridDim.x * blockDim.x;
    long i = (long)blockIdx.x * blockDim.x + threadIdx.x;
    const long tot = (long)N_NODES * OUT_DIM;
    for (long j = i; j < tot; j += stride) out[j] = 0.0f;
    for (long j = i; j < N_NODES; j += stride) { m_enc[j] = ENC_NEG_INF; denom[j] = 0.0f; }
}

// ---------- kernel 2: z = h @ W via V_WMMA_F32_16X16X4_F32 ----------
// Block = 128 threads = 4 waves. Each wave computes one 16x16 tile of z;
// the 4 waves cover all 64 output columns for 16 rows. W staged in LDS (32 KB).
__global__ __launch_bounds__(128)
void gemm_wmma(const float* __restrict__ h, const float* __restrict__ W,
               float* __restrict__ z) {
    __shared__ float Wl[IN_DIM * OUT_DIM];   // 8192 floats = 32 KB (of 320 KB)

    const int tid = threadIdx.x;
    // cooperative 128-bit loads of W into LDS
    for (int i = tid * 4; i < IN_DIM * OUT_DIM; i += 128 * 4) {
        *(float4*)&Wl[i] = *(const float4*)&W[i];
    }
    __syncthreads();

    const int wave    = tid >> 5;        // 0..3 -> 16-column tile
    const int lane    = tid & 31;
    const int l16     = lane & 15;       // M (for A) / N (for B,C,D)
    const int khalf   = (lane >> 4) * 2; // K-pair selector per ISA 16x4 f32 layout
    const int rowBase = blockIdx.x * 16; // N_NODES = 6250 * 16 exactly
    const int colBase = wave * 16;

    const float* hrow = h + (long)(rowBase + l16) * IN_DIM;

    v8f acc = {};
    #pragma unroll 4
    for (int k = 0; k < IN_DIM; k += 4) {
        // A fragment: 16x4 of h. VGPR j of lane holds h[M = lane%16][k + (lane/16)*2 + j]
        v2f a = *(const v2f*)(hrow + k + khalf);
        // B fragment: 4x16 of W. VGPR j of lane holds W[k + (lane/16)*2 + j][N = lane%16]
        v2f b;
        b.x = Wl[(k + khalf + 0) * OUT_DIM + colBase + l16];
        b.y = Wl[(k + khalf + 1) * OUT_DIM + colBase + l16];
        // unconditional speculative prefetch of the next K-block of this h row
        // (prefetch is dropped silently if invalid; no guard -> no branch in loop)
        __builtin_prefetch(hrow + k + 32, 0, 1); // global_prefetch_b8
        // 8 args: (neg_a, A, neg_b, B, c_mod, C, reuse_a, reuse_b)
        acc = __builtin_amdgcn_wmma_f32_16x16x4_f32(
                  false, a, false, b, (short)0, acc, false, false);
    }

    // C/D layout: VGPR v of lane holds z[M = v + (lane/16)*8][N = lane%16]
    float* zt = z + (long)rowBase * OUT_DIM + colBase + l16;
    const int mOff = (lane >> 4) * 8;
    #pragma unroll
    for (int v = 0; v < 8; ++v) {
        zt[(long)(v + mOff) * OUT_DIM] = acc[v];
    }
}

// ---------- kernel 3: s_src / s_dst = z @ a_src, z @ a_dst (wave per node) ----------
__global__ __launch_bounds__(256)
void scores_kernel(const float* __restrict__ z, const float* __restrict__ a,
                   float* __restrict__ s_src, float* __restrict__ s_dst) {
    long wid = ((long)blockIdx.x * blockDim.x + threadIdx.x) >> 5;
    int lane = threadIdx.x & 31;
    if (wid >= N_NODES) return;
    float2 zv = *(const float2*)(z + wid * OUT_DIM + lane * 2);
    float2 as = *(const float2*)(a + lane * 2);
    float2 ad = *(const float2*)(a + OUT_DIM + lane * 2);
    float ps = zv.x * as.x + zv.y * as.y;
    float pd = zv.x * ad.x + zv.y * ad.y;
    #pragma unroll
    for (int off = 16; off > 0; off >>= 1) {
        ps += __shfl_down(ps, off, 32);
        pd += __shfl_down(pd, off, 32);
    }
    if (lane == 0) { s_src[wid] = ps; s_dst[wid] = pd; }
}

// ---------- kernel 4: per-edge leaky-relu attention + segment max ----------
__global__ __launch_bounds__(256)
void edge_att_max(const int* __restrict__ src, const int* __restrict__ dst,
                  const float* __restrict__ s_src, const float* __restrict__ s_dst,
                  float* __restrict__ att, unsigned* __restrict__ m_enc) {
    long e = (long)blockIdx.x * blockDim.x + threadIdx.x;
    if (e >= N_EDGES) return;
    int d = dst[e];
    float v = s_src[src[e]] + s_dst[d];
    float lr = v > 0.0f ? v : 0.01f * v;   // jax leaky_relu default slope
    att[e] = lr;
    atomicMax(&m_enc[d], enc_f(lr));
}

// ---------- kernel 5: decode segment max (-inf -> 0) ----------
__global__ __launch_bounds__(256)
void m_finalize(const unsigned* __restrict__ m_enc, float* __restrict__ m) {
    long i = (long)blockIdx.x * blockDim.x + threadIdx.x;
    if (i >= N_NODES) return;
    float v = dec_f(m_enc[i]);
    m[i] = (m_enc[i] == ENC_NEG_INF) ? 0.0f : v;
}

// ---------- kernel 6: ex = exp(att - m[dst]); denom += ex ----------
__global__ __launch_bounds__(256)
void edge_exp(const int* __restrict__ dst, const float* __restrict__ att,
              const float* __restrict__ m, float* __restrict__ ex,
              float* __restrict__ denom) {
    long e = (long)blockIdx.x * blockDim.x + threadIdx.x;
    if (e >= N_EDGES) return;
    int d = dst[e];
    float x = __expf(att[e] - m[d]);
    ex[e] = x;
    atomAddF32(&denom[d], x);
}

// ---------- kernel 7: out[dst] += (ex/denom) * z[src]  (wave per edge) ----------
__global__ __launch_bounds__(256)
void edge_aggregate(const int* __restrict__ src, const int* __restrict__ dst,
                    const float* __restrict__ z, const float* __restrict__ ex,
                    const float* __restrict__ denom, float* __restrict__ out) {
    long wid = ((long)blockIdx.x * blockDim.x + threadIdx.x) >> 5;
    int lane = threadIdx.x & 31;
    if (wid >= N_EDGES) return;
    int s = src[wid];
    int d = dst[wid];
    float dn = denom[d];
    float alpha = ex[wid] / (dn == 0.0f ? 1.0f : dn);
    float2 zv = *(const float2*)(z + (long)s * OUT_DIM + lane * 2);
    float* o = out + (long)d * OUT_DIM + lane * 2;
    atomAddF32(o,     alpha * zv.x);
    atomAddF32(o + 1, alpha * zv.y);
}

// ---------- host ----------
static inline size_t alignUp(size_t x) { return (x + 255) & ~(size_t)255; }

extern "C" void kernel_launch(void* const* d_in, const int* in_sizes, int n_in,
                              void* d_out, int out_size, void* d_ws, size_t ws_size,
                              hipStream_t stream) {
    const float* h  = (const float*)d_in[0];
    const float* W  = (const float*)d_in[1];
    const float* a  = (const float*)d_in[2];
    const int* src  = (const int*)d_in[3];
    const int* dst  = (const int*)d_in[4];
    float* out      = (float*)d_out;

    // workspace carve-out (~40.4 MB)
    char* ws = (char*)d_ws;
    float*    z     = (float*)ws;    ws += alignUp((size_t)N_NODES * OUT_DIM * 4);
    float*    s_src = (float*)ws;    ws += alignUp((size_t)N_NODES * 4);
    float*    s_dst = (float*)ws;    ws += alignUp((size_t)N_NODES * 4);
    unsigned* m_enc = (unsigned*)ws; ws += alignUp((size_t)N_NODES * 4);
    float*    m     = (float*)ws;    ws += alignUp((size_t)N_NODES * 4);
    float*    denom = (float*)ws;    ws += alignUp((size_t)N_NODES * 4);
    float*    att   = (float*)ws;    ws += alignUp((size_t)N_EDGES * 4);
    float*    ex    = (float*)ws;    ws += alignUp((size_t)N_EDGES * 4);

    init_kernel<<<2048, 256, 0, stream>>>(out, m_enc, denom);

    gemm_wmma<<<N_NODES / 16, 128, 0, stream>>>(h, W, z);                 // 6250 blocks

    long score_threads = (long)N_NODES * 32;
    scores_kernel<<<(unsigned)((score_threads + 255) / 256), 256, 0, stream>>>(z, a, s_src, s_dst);

    edge_att_max<<<(N_EDGES + 255) / 256, 256, 0, stream>>>(src, dst, s_src, s_dst, att, m_enc);

    m_finalize<<<(N_NODES + 255) / 256, 256, 0, stream>>>(m_enc, m);

    edge_exp<<<(N_EDGES + 255) / 256, 256, 0, stream>>>(dst, att, m, ex, denom);

    long agg_threads = (long)N_EDGES * 32;
    edge_aggregate<<<(unsigned)((agg_threads + 255) / 256), 256, 0, stream>>>(src, dst, z, ex, denom, out);
}